// Mk1_37538014167484
// MI455X (gfx1250) — compile-verified
//
#include <hip/hip_runtime.h>

// ---------------------------------------------------------------------------
// Fused 2-layer LSTM stack for gfx1250 (MI455X), fp32 via V_WMMA_F32_16X16X4_F32.
//
// Formulation per timestep (per 16-batch tile, one wave32):
//   z^T[4H x 16] = Wcomb'^T @ [inputs ; h]^T  + bias   (gate-interleaved cols)
// C-layout of z^T: lane = batch (0..15, repeated in lanes 16..31), VGPR v holds
// gate row M = v + 8*(lane>=16). With interleaved gate order M = 4*unit+{i,f,g,o},
// each lane owns whole (i,f,g,o) quadruples -> lane-local state update, and the
// resulting h distribution matches the WMMA B-operand layout for the next step.
//
// Sigmoid on the serial critical path uses gfx1250's v_tanh_f32:
//   sigmoid(x) = 0.5*tanh(0.5*x) + 0.5   (1 trans + 2 VALU ops)
// ---------------------------------------------------------------------------

typedef __attribute__((ext_vector_type(2))) float v2f;
typedef __attribute__((ext_vector_type(8))) float v8f;

#define NB 1024
#define NT 2048

__device__ __forceinline__ float fast_rcp(float x) {
#if __has_builtin(__builtin_amdgcn_rcpf)
    return __builtin_amdgcn_rcpf(x);   // v_rcp_f32
#else
    return 1.0f / x;
#endif
}

__device__ __forceinline__ float sigm(float x) {
#if __has_builtin(__builtin_amdgcn_tanhf)
    return fmaf(0.5f, __builtin_amdgcn_tanhf(0.5f * x), 0.5f);  // v_tanh_f32
#else
    return fast_rcp(1.0f + __expf(-x));                          // v_exp + v_rcp
#endif
}

// z holds gates (i,f,g,o) at literal positions P..P+3 of a v8f accumulator.
#define LSTM_UPD(z, P, h, c)                                     \
    do {                                                         \
        float i_ = sigm((z)[(P)]);                               \
        float f_ = sigm((z)[(P) + 1]);                           \
        float g_ = sigm((z)[(P) + 2]);                           \
        float o_ = sigm((z)[(P) + 3]);                           \
        (c) = f_ * (c) + i_ * g_;                                \
        (h) = o_ * sigm((c));                                    \
    } while (0)

#define WMMA4(A, Bv, C) \
    __builtin_amdgcn_wmma_f32_16x16x4_f32(false, (A), false, (Bv), (short)0, (C), false, false)

__global__ void __launch_bounds__(32)
lstm_stack_kernel(const float* __restrict__ X,
                  const float* __restrict__ Wk1, const float* __restrict__ Wr1,
                  const float* __restrict__ B1,
                  const float* __restrict__ Wk2, const float* __restrict__ Wr2,
                  const float* __restrict__ B2,
                  const float* __restrict__ Wd,  const float* __restrict__ Bd,
                  float* __restrict__ OUT) {
    const int lane = threadIdx.x;
    const int m16  = lane & 15;
    const int hi   = lane >> 4;          // 0 = lanes 0-15, 1 = lanes 16-31
    const bool lo  = (hi == 0);
    const int b0   = blockIdx.x * 16;    // this wave owns batch rows b0..b0+15

    // ---- Preload A operands (gate-interleaved, transposed weights) ----------
    // LSTM1: A1[M=16][K=8], M<12 used (unit=M/4, gate=M%4), K<5 = [x0,x1,h0,h1,h2]
    v2f a1[2];
#pragma unroll
    for (int c = 0; c < 2; ++c) {
#pragma unroll
        for (int v = 0; v < 2; ++v) {
            const int K = 4 * c + v + 2 * hi;
            float val = 0.0f;
            if (m16 < 12 && K < 5) {
                const int col = (m16 & 3) * 3 + (m16 >> 2);   // gate*H1 + unit
                val = (K < 2) ? Wk1[K * 12 + col] : Wr1[(K - 2) * 12 + col];
            }
            a1[c][v] = val;
        }
    }
    // LSTM2: A2[M=48][K=20], M<36 used, K<18 = [y4(9) ; h2(9)]
    v2f a2[3][5];
#pragma unroll
    for (int m = 0; m < 3; ++m) {
#pragma unroll
        for (int c = 0; c < 5; ++c) {
#pragma unroll
            for (int v = 0; v < 2; ++v) {
                const int K  = 4 * c + v + 2 * hi;
                const int Mg = 16 * m + m16;
                float val = 0.0f;
                if (Mg < 36 && K < 18) {
                    const int col = (Mg & 3) * 9 + (Mg >> 2); // gate*H2 + unit
                    val = (K < 9) ? Wk2[K * 36 + col] : Wr2[(K - 9) * 36 + col];
                }
                a2[m][c][v] = val;
            }
        }
    }
    // Biases in C-layout (used as WMMA C initializer)
    v8f bc1;
#pragma unroll
    for (int v = 0; v < 8; ++v) {
        const int Mg = v + 8 * hi;
        bc1[v] = (Mg < 12) ? B1[(Mg & 3) * 3 + (Mg >> 2)] : 0.0f;
    }
    v8f bc2[3];
#pragma unroll
    for (int m = 0; m < 3; ++m) {
#pragma unroll
        for (int v = 0; v < 8; ++v) {
            const int Mg = 16 * m + v + 8 * hi;
            bc2[m][v] = (Mg < 36) ? B2[(Mg & 3) * 9 + (Mg >> 2)] : 0.0f;
        }
    }
    // Dense head weights, lane-half dependent unit mapping:
    // lanes 0-15 hold h2 units {0,1,4,5,8}; lanes 16-31 hold {2,3,6,7}.
    const int umap_lo[5] = {0, 1, 4, 5, 8};
    const int umap_hi[5] = {2, 3, 6, 7, -1};
    float wdl[5][3];
#pragma unroll
    for (int k = 0; k < 5; ++k) {
        const int u = lo ? umap_lo[k] : umap_hi[k];
#pragma unroll
        for (int d = 0; d < 3; ++d) wdl[k][d] = (u >= 0) ? Wd[u * 3 + d] : 0.0f;
    }
    const float bd0 = Bd[0], bd1 = Bd[1], bd2 = Bd[2];

    // ---- Recurrent state (lane-local, see layout comment above) -------------
    // LSTM1 per branch: lo lanes hold units {0,1} in (hB0,hB1); hi lanes hold
    // unit {2} in hB0 (hB1 is padding rows, never consumed).
    float hB0[3] = {0, 0, 0}, hB1[3] = {0, 0, 0};
    float cB0[3] = {0, 0, 0}, cB1[3] = {0, 0, 0};
    // LSTM2: lo lanes: units {0,1,4,5,8} in hA[0..4]; hi lanes: {2,3,6,7} in hA[0..3].
    float hA[5] = {0, 0, 0, 0, 0}, cA[5] = {0, 0, 0, 0, 0};

    // Both lane halves read the SAME batch row (b0 + m16): identical addresses
    // coalesce in L0/L2, and it keeps EXEC fully populated (no per-step branch).
    const float* xrow = X + ((size_t)(b0 + m16) * NT) * 6;

#pragma unroll 1
    for (int t = 0; t < NT; ++t) {
        // ---- load x[b, t, 0..5] (all lanes; hi lanes mirror lo lanes) -------
        const float* xp = xrow + (size_t)t * 6;
        float x0[3], x1[3];
        x0[0] = xp[0]; x1[0] = xp[1];
        x0[1] = xp[2]; x1[1] = xp[3];
        x0[2] = xp[4]; x1[2] = xp[5];
        __builtin_prefetch(xp + 6, 0, 1);   // next timestep's row (speculative)

        // ---- LSTM1, 3 shared-weight branches --------------------------------
        // B operand K layout: K0,K1 = x(2 feats); K2,K3,K4 = h(3); K5..7 = 0.
#pragma unroll
        for (int br = 0; br < 3; ++br) {
            const float t0 = __shfl_xor(hB0[br], 16, 32); // lo<->hi partner values
            const float t1 = __shfl_xor(hB1[br], 16, 32);
            v2f Bk0, Bk1;
            Bk0[0] = lo ? x0[br] : t0;   // K0 (x0) | K2 (h0)
            Bk0[1] = lo ? x1[br] : t1;   // K1 (x1) | K3 (h1)
            Bk1[0] = lo ? t0 : 0.0f;     // K4 (h2) | K6 (pad)
            Bk1[1] = 0.0f;               // K5/K7 pad (A cols are zero anyway)
            v8f z = bc1;
            z = WMMA4(a1[0], Bk0, z);
            z = WMMA4(a1[1], Bk1, z);
            LSTM_UPD(z, 0, hB0[br], cB0[br]); // lo: unit0 | hi: unit2
            LSTM_UPD(z, 4, hB1[br], cB1[br]); // lo: unit1 | hi: pad (unused)
        }

        // ---- LSTM2: K = [y4(t) = concat h1 branches (9) ; h2(t-1) (9)] ------
        const float p01 = __shfl_xor(hB0[1], 16, 32);
        const float p02 = __shfl_xor(hB0[2], 16, 32);
        const float p12 = __shfl_xor(hB1[2], 16, 32);
        const float pA1 = __shfl_xor(hA[1], 16, 32);
        const float pA3 = __shfl_xor(hA[3], 16, 32);
        v2f Bc[5];
        Bc[0][0] = hB0[0];                  // K0  b0u0 | K2  b0u2   (both resident!)
        Bc[0][1] = lo ? hB1[0] : p01;       // K1  b0u1 | K3  b1u0
        Bc[1][0] = lo ? hB1[1] : p02;       // K4  b1u1 | K6  b2u0
        Bc[1][1] = lo ? p01 : p12;          // K5  b1u2 | K7  b2u1
        Bc[2][0] = lo ? p02 : pA1;          // K8  b2u2 | K10 h2u1
        Bc[2][1] = hA[0];                   // K9  h2u0 | K11 h2u2   (both resident!)
        Bc[3][0] = lo ? pA1 : pA3;          // K12 h2u3 | K14 h2u5
        Bc[3][1] = hA[2];                   // K13 h2u4 | K15 h2u6   (both resident!)
        Bc[4][0] = lo ? pA3 : 0.0f;         // K16 h2u7 | K18 pad
        Bc[4][1] = lo ? hA[4] : 0.0f;       // K17 h2u8 | K19 pad

        v8f z0 = bc2[0], z1 = bc2[1], z2 = bc2[2];
#pragma unroll
        for (int c = 0; c < 5; ++c) {
            z0 = WMMA4(a2[0][c], Bc[c], z0);
            z1 = WMMA4(a2[1][c], Bc[c], z1);
            z2 = WMMA4(a2[2][c], Bc[c], z2);
        }
        LSTM_UPD(z0, 0, hA[0], cA[0]);  // lo: u0 | hi: u2
        LSTM_UPD(z0, 4, hA[1], cA[1]);  // lo: u1 | hi: u3
        LSTM_UPD(z1, 0, hA[2], cA[2]);  // lo: u4 | hi: u6
        LSTM_UPD(z1, 4, hA[3], cA[3]);  // lo: u5 | hi: u7
        LSTM_UPD(z2, 0, hA[4], cA[4]);  // lo: u8 | hi: pad (wdl[4]=0, Bc hi=0)

        // ---- dense head: y = sigmoid(h2 @ Wd + bd), half-sum across lanes ---
        float s0 = 0.0f, s1 = 0.0f, s2 = 0.0f;
#pragma unroll
        for (int k = 0; k < 5; ++k) {
            s0 = fmaf(hA[k], wdl[k][0], s0);
            s1 = fmaf(hA[k], wdl[k][1], s1);
            s2 = fmaf(hA[k], wdl[k][2], s2);
        }
        s0 += __shfl_xor(s0, 16, 32);
        s1 += __shfl_xor(s1, 16, 32);
        s2 += __shfl_xor(s2, 16, 32);
        if (lo) {
            float* op = OUT + ((size_t)(b0 + m16) * NT + t) * 3;
            op[0] = sigm(s0 + bd0);
            op[1] = sigm(s1 + bd1);
            op[2] = sigm(s2 + bd2);
        }
    }
}

extern "C" void kernel_launch(void* const* d_in, const int* in_sizes, int n_in,
                              void* d_out, int out_size, void* d_ws, size_t ws_size,
                              hipStream_t stream) {
    (void)in_sizes; (void)n_in; (void)out_size; (void)d_ws; (void)ws_size;
    const float* x   = (const float*)d_in[0];
    const float* wk1 = (const float*)d_in[1];
    const float* wr1 = (const float*)d_in[2];
    const float* b1  = (const float*)d_in[3];
    const float* wk2 = (const float*)d_in[4];
    const float* wr2 = (const float*)d_in[5];
    const float* b2  = (const float*)d_in[6];
    const float* wd  = (const float*)d_in[7];
    const float* bd  = (const float*)d_in[8];
    float* out = (float*)d_out;

    dim3 grid(NB / 16);   // 64 batch tiles of 16 rows, one wave32 each
    dim3 block(32);
    lstm_stack_kernel<<<grid, block, 0, stream>>>(x, wk1, wr1, b1, wk2, wr2, b2,
                                                  wd, bd, out);
}